// GraphVAE_19971597926822
// MI455X (gfx1250) — compile-verified
//
#include <hip/hip_runtime.h>

typedef __attribute__((ext_vector_type(2))) float v2f;
typedef __attribute__((ext_vector_type(8))) float v8f;

#define HIDD 128
#define ZD   64
#define HEADS 4

// ---------------------------------------------------------------------------
// Generic row-major GEMM via V_WMMA_F32_16X16X4_F32. One wave32 per 16x16 tile.
// C[M,Nn] = act(A[M,K] @ B[K,Nn] + bias + resid)
// Requires M%16==0, Nn%16==0, K%4==0.
// A layout (16x4): lane<16 -> row=lane, K={k,k+1}; lane>=16 -> row=lane-16, K={k+2,k+3}
// B layout (4x16): v[0] holds row k+2*half, v[1] row k+2*half+1, col = lane%16
// C layout: v[i] -> row = i + 8*half, col = lane%16
// ---------------------------------------------------------------------------
__global__ __launch_bounds__(256) void gemm_f32_wmma(
    const float* __restrict__ A, const float* __restrict__ B,
    const float* __restrict__ bias, const float* __restrict__ resid,
    float* __restrict__ C, int M, int Nn, int K, int doRelu)
{
    const int lane = threadIdx.x & 31;
    const int wave = threadIdx.x >> 5;
    const int tilesN = Nn >> 4;
    const int tilesM = M >> 4;
    const int tile = blockIdx.x * 8 + wave;
    if (tile >= tilesM * tilesN) return;            // wave-uniform: EXEC stays all-1
    const int tm = tile / tilesN, tn = tile % tilesN;
    const int half = lane >> 4;
    const int l16  = lane & 15;

    const float* Arow = A + (size_t)(tm * 16 + l16) * K;
    const float* Bcol = B + (tn * 16 + l16);

    v8f acc = {};
    for (int k = 0; k < K; k += 4) {
        v2f a = *(const v2f*)(Arow + k + 2 * half);
        v2f b;
        b.x = Bcol[(size_t)(k + 2 * half)     * Nn];
        b.y = Bcol[(size_t)(k + 2 * half + 1) * Nn];
        acc = __builtin_amdgcn_wmma_f32_16x16x4_f32(false, a, false, b,
                                                    (short)0, acc, false, false);
    }

    const float bv = bias ? bias[tn * 16 + l16] : 0.0f;
#pragma unroll
    for (int i = 0; i < 8; ++i) {
        const int row = tm * 16 + i + 8 * half;
        const size_t idx = (size_t)row * Nn + tn * 16 + l16;
        float v = acc[i] + bv;
        if (resid) v += resid[idx];
        if (doRelu) v = fmaxf(v, 0.0f);
        C[idx] = v;
    }
}

// ---------------------------------------------------------------------------
// logits = D @ D^T, D is [M, 64] f32.  B[k][n] = D[n][k] so both operands read
// the same row-major layout. Fully unrolled: 16 WMMAs back-to-back per tile.
// ---------------------------------------------------------------------------
__global__ __launch_bounds__(256) void aat_f32_wmma(
    const float* __restrict__ D, float* __restrict__ out, int M)
{
    const int lane = threadIdx.x & 31;
    const int wave = threadIdx.x >> 5;
    const int tilesN = M >> 4;
    const int tile = blockIdx.x * 8 + wave;
    if (tile >= tilesN * tilesN) return;
    const int tm = tile / tilesN, tn = tile % tilesN;
    const int half = lane >> 4;
    const int l16  = lane & 15;

    const float* Arow = D + (size_t)(tm * 16 + l16) * ZD + 2 * half;
    const float* Brow = D + (size_t)(tn * 16 + l16) * ZD + 2 * half;

    v8f acc = {};
#pragma unroll
    for (int k = 0; k < ZD; k += 4) {
        v2f a = *(const v2f*)(Arow + k);
        v2f b = *(const v2f*)(Brow + k);
        acc = __builtin_amdgcn_wmma_f32_16x16x4_f32(false, a, false, b,
                                                    (short)0, acc, false, false);
    }
#pragma unroll
    for (int i = 0; i < 8; ++i) {
        const int row = tm * 16 + i + 8 * half;
        out[(size_t)row * M + tn * 16 + l16] = acc[i];
    }
}

// ---------------------------------------------------------------------------
// Graph / elementwise kernels
// ---------------------------------------------------------------------------
__device__ inline void atomicMaxF(float* addr, float val) {
    unsigned int* ua = (unsigned int*)addr;
    unsigned int old = *ua;
    while (__uint_as_float(old) < val) {
        unsigned int assumed = old;
        old = atomicCAS(ua, assumed, __float_as_uint(val));
        if (old == assumed) break;
    }
}

__global__ void k_init_deg(float* deg, int n) {
    int i = blockIdx.x * blockDim.x + threadIdx.x;
    if (i < n) deg[i] = 1.0f;                       // self loop
}

__global__ void k_deg_edges(const int* __restrict__ ei, float* deg, int E) {
    int e = blockIdx.x * blockDim.x + threadIdx.x;
    if (e < E) atomicAdd(&deg[ei[E + e]], 1.0f);
}

__global__ void k_dinv(const float* __restrict__ deg, float* dinv, int n) {
    int i = blockIdx.x * blockDim.x + threadIdx.x;
    if (i < n) dinv[i] = rsqrtf(deg[i]);            // deg >= 1 always
}

// self-loop contribution initializes acc (one writer per element, no atomics)
__global__ void k_gcn_self(const float* __restrict__ h0, const float* __restrict__ dinv,
                           float* acc, int n) {
    int idx = blockIdx.x * blockDim.x + threadIdx.x;
    if (idx < n * HIDD) {
        int node = idx >> 7;
        float d = dinv[node];
        acc[idx] = h0[idx] * d * d;
    }
}

// one wave per edge: 32 lanes x 4 floats = 128
__global__ void k_gcn_edges(const int* __restrict__ ei, const float* __restrict__ h0,
                            const float* __restrict__ dinv, float* acc, int E) {
    int gid  = blockIdx.x * (blockDim.x >> 5) + (threadIdx.x >> 5);
    int lane = threadIdx.x & 31;
    if (gid >= E) return;
    int s = ei[gid], d = ei[E + gid];
    float c = dinv[s] * dinv[d];
    const float* hs = h0 + (size_t)s * HIDD;
    float* ad = acc + (size_t)d * HIDD;
    for (int j = lane; j < HIDD; j += 32)
        atomicAdd(&ad[j], hs[j] * c);
}

__global__ void k_bias_relu(const float* __restrict__ in, const float* __restrict__ b,
                            float* out, int total, int cols) {
    int idx = blockIdx.x * blockDim.x + threadIdx.x;
    if (idx < total) out[idx] = fmaxf(in[idx] + b[idx % cols], 0.0f);
}

__global__ void k_att_scores(const float* __restrict__ g, const float* __restrict__ att_s,
                             const float* __restrict__ att_d, float* a_src, float* a_dst, int n) {
    int idx = blockIdx.x * blockDim.x + threadIdx.x;
    if (idx >= n * HEADS) return;
    int node = idx >> 2, h = idx & 3;
    const float* gp = g + (size_t)node * (HEADS * HIDD) + h * HIDD;
    const float* as = att_s + h * HIDD;
    const float* ad = att_d + h * HIDD;
    float s1 = 0.0f, s2 = 0.0f;
    for (int j = 0; j < HIDD; ++j) { float v = gp[j]; s1 += v * as[j]; s2 += v * ad[j]; }
    a_src[idx] = s1; a_dst[idx] = s2;
}

__global__ void k_init_gat(float* emax, float* den, float* agg, int n) {
    int idx = blockIdx.x * blockDim.x + threadIdx.x;
    if (idx < n * (HEADS * HIDD)) agg[idx] = 0.0f;
    if (idx < n * HEADS) { emax[idx] = -__builtin_inff(); den[idx] = 0.0f; }
}

__global__ void k_gat_edge_max(const int* __restrict__ ei, const float* __restrict__ a_src,
                               const float* __restrict__ a_dst, float* ebuf, float* emax,
                               int E, int n) {
    int idx = blockIdx.x * blockDim.x + threadIdx.x;
    if (idx >= (E + n) * HEADS) return;
    int e = idx >> 2, h = idx & 3, s, d;
    if (e < E) { s = ei[e]; d = ei[E + e]; } else { s = d = e - E; }
    float v = a_src[s * HEADS + h] + a_dst[d * HEADS + h];
    v = v > 0.0f ? v : 0.2f * v;                    // leaky_relu 0.2
    ebuf[idx] = v;
    atomicMaxF(&emax[d * HEADS + h], v);
}

__global__ void k_gat_edge_exp(const int* __restrict__ ei, float* ebuf,
                               const float* __restrict__ emax, float* den, int E, int n) {
    int idx = blockIdx.x * blockDim.x + threadIdx.x;
    if (idx >= (E + n) * HEADS) return;
    int e = idx >> 2, h = idx & 3;
    int d = (e < E) ? ei[E + e] : (e - E);
    float ex = expf(ebuf[idx] - emax[d * HEADS + h]);
    ebuf[idx] = ex;
    atomicAdd(&den[d * HEADS + h], ex);
}

// thread per (edge, i) with i in [0,512): h = i/128
__global__ void k_gat_edge_agg(const int* __restrict__ ei, const float* __restrict__ ebuf,
                               const float* __restrict__ den, const float* __restrict__ g,
                               float* agg, int E, int n) {
    int idx = blockIdx.x * blockDim.x + threadIdx.x;
    int total = (E + n) * (HEADS * HIDD);
    if (idx >= total) return;
    int e = idx >> 9, i = idx & 511, h = i >> 7, s, d;
    if (e < E) { s = ei[e]; d = ei[E + e]; } else { s = d = e - E; }
    float alpha = ebuf[e * HEADS + h] / den[d * HEADS + h];
    atomicAdd(&agg[(size_t)d * 512 + i], alpha * g[(size_t)s * 512 + i]);
}

__global__ void k_gat_reduce(const float* __restrict__ agg, const float* __restrict__ gat_b,
                             float* h2, int n) {
    int idx = blockIdx.x * blockDim.x + threadIdx.x;
    if (idx >= n * HIDD) return;
    int node = idx >> 7, j = idx & 127;
    const float* a = agg + (size_t)node * 512;
    float v = 0.25f * (a[j] + a[128 + j] + a[256 + j] + a[384 + j]) + gat_b[j];
    h2[idx] = fmaxf(v, 0.0f);
}

__global__ void k_reparam(const float* __restrict__ mu, const float* __restrict__ lv,
                          const float* __restrict__ eps, float* z, int total) {
    int idx = blockIdx.x * blockDim.x + threadIdx.x;
    if (idx >= total) return;
    float l = fminf(fmaxf(lv[idx], -10.0f), 10.0f);
    z[idx] = mu[idx] + eps[idx] * expf(0.5f * l);
}

// ---------------------------------------------------------------------------
// Host launch
// ---------------------------------------------------------------------------
static inline int cdiv(int a, int b) { return (a + b - 1) / b; }

extern "C" void kernel_launch(void* const* d_in, const int* in_sizes, int n_in,
                              void* d_out, int out_size, void* d_ws, size_t ws_size,
                              hipStream_t stream) {
    const float* x        = (const float*)d_in[0];
    const int*   ei       = (const int*)  d_in[1];
    const float* eps      = (const float*)d_in[2];
    const float* gcn_w    = (const float*)d_in[3];
    const float* gcn_b    = (const float*)d_in[4];
    const float* gat_w    = (const float*)d_in[5];
    const float* gat_as   = (const float*)d_in[6];
    const float* gat_ad   = (const float*)d_in[7];
    const float* gat_b    = (const float*)d_in[8];
    const float* mlp_w1   = (const float*)d_in[9];
    const float* mlp_b1   = (const float*)d_in[10];
    const float* mlp_w2   = (const float*)d_in[11];
    const float* mlp_b2   = (const float*)d_in[12];
    const float* mu_w     = (const float*)d_in[13];
    const float* mu_b     = (const float*)d_in[14];
    const float* lv_w     = (const float*)d_in[15];
    const float* lv_b     = (const float*)d_in[16];
    const float* dec_w1   = (const float*)d_in[17];
    const float* dec_b1   = (const float*)d_in[18];
    const float* dec_w2   = (const float*)d_in[19];
    const float* dec_b2   = (const float*)d_in[20];

    const int n  = in_sizes[0] / HIDD;        // 10000
    const int E  = in_sizes[1] / 2;           // 320000
    float* out   = (float*)d_out;
    float* muO   = out + (size_t)n * n;       // outputs: logits, mu, logvar
    float* lvO   = muO + (size_t)n * ZD;

    float* ws = (float*)d_ws;
    size_t o = 0;
    float* h0   = ws + o; o += (size_t)n * HIDD;
    float* deg  = ws + o; o += n;
    float* dinv = ws + o; o += n;
    float* acc  = ws + o; o += (size_t)n * HIDD;
    float* hg   = ws + o; o += (size_t)n * HIDD;         // post-GCN activations
    float* g    = ws + o; o += (size_t)n * HEADS * HIDD;
    float* asr  = ws + o; o += (size_t)n * HEADS;
    float* adt  = ws + o; o += (size_t)n * HEADS;
    float* emax = ws + o; o += (size_t)n * HEADS;
    float* den  = ws + o; o += (size_t)n * HEADS;
    float* ebuf = ws + o; o += (size_t)(E + n) * HEADS;
    float* agg  = ws + o; o += (size_t)n * HEADS * HIDD;
    float* h2   = ws + o; o += (size_t)n * HIDD;
    float* h3   = ws + o; o += (size_t)n * HIDD;
    float* h4   = ws + o; o += (size_t)n * HIDD;
    float* z    = ws + o; o += (size_t)n * ZD;
    float* dd1  = ws + o; o += (size_t)n * HIDD;
    float* dfin = ws + o; o += (size_t)n * ZD;

    const int T = 256;
    auto gemmGrid = [](int M, int Nn) { return cdiv((M / 16) * (Nn / 16), 8); };

    // ---- GCN ----
    gemm_f32_wmma<<<gemmGrid(n, HIDD), T, 0, stream>>>(x, gcn_w, nullptr, nullptr,
                                                       h0, n, HIDD, HIDD, 0);
    k_init_deg<<<cdiv(n, T), T, 0, stream>>>(deg, n);
    k_deg_edges<<<cdiv(E, T), T, 0, stream>>>(ei, deg, E);
    k_dinv<<<cdiv(n, T), T, 0, stream>>>(deg, dinv, n);
    k_gcn_self<<<cdiv(n * HIDD, T), T, 0, stream>>>(h0, dinv, acc, n);
    k_gcn_edges<<<cdiv(E, 8), T, 0, stream>>>(ei, h0, dinv, acc, E);
    k_bias_relu<<<cdiv(n * HIDD, T), T, 0, stream>>>(acc, gcn_b, hg, n * HIDD, HIDD);

    // ---- GAT ----
    gemm_f32_wmma<<<gemmGrid(n, HEADS * HIDD), T, 0, stream>>>(hg, gat_w, nullptr, nullptr,
                                                               g, n, HEADS * HIDD, HIDD, 0);
    k_att_scores<<<cdiv(n * HEADS, T), T, 0, stream>>>(g, gat_as, gat_ad, asr, adt, n);
    k_init_gat<<<cdiv(n * HEADS * HIDD, T), T, 0, stream>>>(emax, den, agg, n);
    k_gat_edge_max<<<cdiv((E + n) * HEADS, T), T, 0, stream>>>(ei, asr, adt, ebuf, emax, E, n);
    k_gat_edge_exp<<<cdiv((E + n) * HEADS, T), T, 0, stream>>>(ei, ebuf, emax, den, E, n);
    k_gat_edge_agg<<<cdiv((E + n) * HEADS * HIDD, T), T, 0, stream>>>(ei, ebuf, den, g, agg, E, n);
    k_gat_reduce<<<cdiv(n * HIDD, T), T, 0, stream>>>(agg, gat_b, h2, n);

    // ---- MLP encoder ----
    gemm_f32_wmma<<<gemmGrid(n, HIDD), T, 0, stream>>>(h2, mlp_w1, mlp_b1, nullptr,
                                                       h3, n, HIDD, HIDD, 1);
    gemm_f32_wmma<<<gemmGrid(n, HIDD), T, 0, stream>>>(h3, mlp_w2, mlp_b2, nullptr,
                                                       h4, n, HIDD, HIDD, 1);
    gemm_f32_wmma<<<gemmGrid(n, ZD), T, 0, stream>>>(h4, mu_w, mu_b, nullptr,
                                                     muO, n, ZD, HIDD, 0);
    gemm_f32_wmma<<<gemmGrid(n, ZD), T, 0, stream>>>(h4, lv_w, lv_b, nullptr,
                                                     lvO, n, ZD, HIDD, 0);

    // ---- reparameterize ----
    k_reparam<<<cdiv(n * ZD, T), T, 0, stream>>>(muO, lvO, eps, z, n * ZD);

    // ---- decoder ----
    gemm_f32_wmma<<<gemmGrid(n, HIDD), T, 0, stream>>>(z, dec_w1, dec_b1, nullptr,
                                                       dd1, n, HIDD, ZD, 1);
    gemm_f32_wmma<<<gemmGrid(n, ZD), T, 0, stream>>>(dd1, dec_w2, dec_b2, z,
                                                     dfin, n, ZD, HIDD, 0);

    // ---- logits = d @ d^T (dominant: 12.8 GFLOP, 400 MB store) ----
    aat_f32_wmma<<<gemmGrid(n, n), T, 0, stream>>>(dfin, out, n);
}